// DCModule_39719857554087
// MI455X (gfx1250) — compile-verified
//
#include <hip/hip_runtime.h>
#include <cstdint>
#include <cstddef>

// ---------------------------------------------------------------------------
// Windowed argmin/argmax pooling refine (WS=3, SS=2) on 4096x4096 f32.
// out[r][c] = sel[min(r/2,2046)][min(c/2,2046)] for r,c < 4095; passthrough on
// the last row/col. Memory-bound (~320MB traffic -> ~14us @ 23.3 TB/s).
// Uses CDNA5 async global->LDS data mover (ASYNCcnt) for tile staging.
// ---------------------------------------------------------------------------

#define H 4096
#define W 4096
#define NWIN 2047          // (4096-3)/2 + 1 windows per dimension
#define TILE 64            // output pixels per block per dimension
#define NWT 32             // windows per block per dimension
#define LROWS 65           // input halo rows per tile
#define LSTRIDE 68         // LDS row stride in floats (16B aligned, bank-friendly)
#define LSIZE (LROWS * LSTRIDE)   // 4420 floats per staged array

typedef float f32x2 __attribute__((ext_vector_type(2)));

// Async DMA of 16 bytes from global memory into LDS (per lane).
// gfx1250: tracked by ASYNCcnt; LDS byte address in a VGPR, 64-bit global
// address in a VGPR pair, SADDR = off.
__device__ __forceinline__ void async_ld_b128(uint32_t lds_byte_off, const float* g) {
  asm volatile("global_load_async_to_lds_b128 %0, %1, off"
               :
               : "v"(lds_byte_off), "v"((unsigned long long)(uintptr_t)g)
               : "memory");
}

__device__ __forceinline__ void wait_asynccnt0() {
  asm volatile("s_wait_asynccnt 0x0" ::: "memory");
}

__global__ __launch_bounds__(256)
void refine_pool_kernel(const float* __restrict__ anchor,
                        const float* __restrict__ positive,
                        const float* __restrict__ negative,
                        float* __restrict__ outP,
                        float* __restrict__ outN) {
  __shared__ __align__(16) float sm[3 * LSIZE];   // [anchor | positive | negative]

  const int t  = threadIdx.x;
  const int bx = blockIdx.x, by = blockIdx.y;
  const int gr0 = by * TILE, gc0 = bx * TILE;

  // Edge tiles: clamp the (row, chunk) loader coordinates so every lane always
  // issues an in-bounds async load with uniform EXEC (duplicates are benign:
  // same source -> same LDS slot -> same data).
  const int maxRow = (by == 63) ? 63 : 64;   // halo row 64 doesn't exist on last tile
  const int maxCh  = (bx == 63) ? 15 : 16;   // chunk 16 (cols 64..67) OOB on last tile

  // Low 32 bits of the flat shared-memory address are the LDS byte offset.
  const uint32_t lbase = (uint32_t)(uintptr_t)(&sm[0]);

  const float* bases[3] = {anchor, positive, negative};
  #pragma unroll
  for (int a = 0; a < 3; ++a) {
    const float* gb = bases[a];
    const uint32_t lb = lbase + (uint32_t)(a * LSIZE * 4);
    #pragma unroll
    for (int k = 0; k < 5; ++k) {
      int cidx = t + k * 256;                 // 0..1279, need 65*17 = 1105 chunks
      int row  = cidx / 17;
      int ch   = cidx - row * 17;
      row = (row < maxRow) ? row : maxRow;
      ch  = (ch  < maxCh)  ? ch  : maxCh;
      const float* g = gb + (size_t)(gr0 + row) * W + (gc0 + ch * 4);
      async_ld_b128(lb + (uint32_t)((row * LSTRIDE + ch * 4) * 4), g);
    }
  }
  wait_asynccnt0();
  __syncthreads();

  // ---- compute 32x32 windows; each thread handles 4 windows ----
  #pragma unroll
  for (int w = 0; w < 4; ++w) {
    const int widx = t + w * 256;             // 0..1023
    const int wi = widx >> 5;
    const int wj = widx & 31;
    const int gi = by * NWT + wi;
    const int gj = bx * NWT + wj;
    if (gi >= NWIN || gj >= NWIN) continue;   // window 2047 doesn't exist

    const int base = (wi * 2) * LSTRIDE + wj * 2;
    float bdp = 3.4e38f, selp = 0.0f;         // argmin |a-p|, first occurrence wins
    float bdn = -1.0f,   seln = 0.0f;         // argmax |a-n|, first occurrence wins
    #pragma unroll
    for (int dr = 0; dr < 3; ++dr) {
      #pragma unroll
      for (int dc = 0; dc < 3; ++dc) {
        const int o = base + dr * LSTRIDE + dc;
        const float av = sm[o];
        const float pv = sm[LSIZE + o];
        const float nv = sm[2 * LSIZE + o];
        const float dp = __builtin_fabsf(av - pv);
        const float dn = __builtin_fabsf(av - nv);
        if (dp < bdp) { bdp = dp; selp = pv; } // strict: row-major first idx wins
        if (dn > bdn) { bdn = dn; seln = nv; }
      }
    }

    const int or0 = gi * 2, oc0 = gj * 2;
    const int nr = (gi == NWIN - 1) ? 3 : 2;  // window 2046 also owns row 4094
    if (gj != NWIN - 1) {
      const f32x2 vp = {selp, selp};
      const f32x2 vn = {seln, seln};
      #pragma unroll
      for (int rr = 0; rr < 3; ++rr) {
        if (rr < nr) {
          __builtin_nontemporal_store(vp, (f32x2*)(outP + (size_t)(or0 + rr) * W + oc0));
          __builtin_nontemporal_store(vn, (f32x2*)(outN + (size_t)(or0 + rr) * W + oc0));
        }
      }
    } else {                                   // window 2046 also owns col 4094
      for (int rr = 0; rr < nr; ++rr)
        for (int cc = 0; cc < 3; ++cc) {
          __builtin_nontemporal_store(selp, outP + (size_t)(or0 + rr) * W + oc0 + cc);
          __builtin_nontemporal_store(seln, outN + (size_t)(or0 + rr) * W + oc0 + cc);
        }
    }
  }

  // ---- passthrough: last output row/col copy comp unchanged ----
  if (by == 63 && t < TILE) {                  // row 4095 (incl. corner 4095,4095)
    const size_t o = (size_t)(H - 1) * W + gc0 + t;
    __builtin_nontemporal_store(sm[LSIZE     + 63 * LSTRIDE + t], outP + o);
    __builtin_nontemporal_store(sm[2 * LSIZE + 63 * LSTRIDE + t], outN + o);
  }
  if (bx == 63 && t < TILE) {                  // col 4095, excluding the corner
    const int r = gr0 + t;
    if (r != H - 1) {
      const size_t o = (size_t)r * W + (W - 1);
      __builtin_nontemporal_store(sm[LSIZE     + t * LSTRIDE + 63], outP + o);
      __builtin_nontemporal_store(sm[2 * LSIZE + t * LSTRIDE + 63], outN + o);
    }
  }
}

extern "C" void kernel_launch(void* const* d_in, const int* in_sizes, int n_in,
                              void* d_out, int out_size, void* d_ws, size_t ws_size,
                              hipStream_t stream) {
  (void)in_sizes; (void)n_in; (void)out_size; (void)d_ws; (void)ws_size;
  const float* anchor   = (const float*)d_in[0];
  const float* positive = (const float*)d_in[1];
  const float* negative = (const float*)d_in[2];
  float* outP = (float*)d_out;
  float* outN = outP + (size_t)H * W;

  dim3 grid(W / TILE, H / TILE);   // 64 x 64 tiles
  dim3 block(256);                 // 8 wave32
  refine_pool_kernel<<<grid, block, 0, stream>>>(anchor, positive, negative, outP, outN);
}